// CEKT_18193481466012
// MI455X (gfx1250) — compile-verified
//
#include <hip/hip_runtime.h>
#include <math.h>

// CDNA5 fp32 WMMA operand types
typedef __attribute__((ext_vector_type(2))) float v2f;
typedef __attribute__((ext_vector_type(8))) float v8f;

#define N_ROWS 4096
#define SA 324   // LDS stride for [e|r|h_prev] tile (324 % 64 == 4 -> conflict-free)
#define SX 132   // LDS stride for x tile
#define SH 132   // LDS stride for h_t tile

// One block: 16 rows x 128 cols of output. 8 waves; wave w -> cols [16w,16w+16).
// GEMM1: [e|r|h_prev](16x320) x {Wq,Wh}(320x128) -> g, cand -> h_t (kept in LDS)
// GEMM2: [h_t|e|x](16x384)    x  Wp(384x128)     -> sigmoid -> y
__global__ __launch_bounds__(256) void cekt_fused_kernel(
    const float* __restrict__ x,
    const float* __restrict__ e_t,
    const float* __restrict__ r_t,
    const float* __restrict__ prev_h,
    const float* __restrict__ Wq_w, const float* __restrict__ Wq_b,
    const float* __restrict__ Wh_w, const float* __restrict__ Wh_b,
    const float* __restrict__ Wp_w, const float* __restrict__ Wp_b,
    float* __restrict__ out)
{
    __shared__ float comb[16 * SA];  // cols 0..127 = e_t, 128..191 = r_t, 192..319 = prev_h
    __shared__ float xs[16 * SX];
    __shared__ float ht[16 * SH];

    const int tid     = threadIdx.x;
    const int lane    = tid & 31;
    const int wave    = tid >> 5;
    const int half    = lane >> 4;   // 0: lanes 0-15, 1: lanes 16-31
    const int lrow    = lane & 15;
    const int colbase = wave * 16;
    const int row0    = blockIdx.x * 16;

    // Warm L2 with the (block-shared) weight matrices: emits global_prefetch_b8.
    for (int i = tid; i < (320 * 128) / 64; i += 256) {
        __builtin_prefetch(Wq_w + i * 64, 0, 0);
        __builtin_prefetch(Wh_w + i * 64, 0, 0);
    }
    for (int i = tid; i < (384 * 128) / 64; i += 256)
        __builtin_prefetch(Wp_w + i * 64, 0, 0);

    // Stage activation tiles into LDS.
    for (int idx = tid; idx < 16 * 128; idx += 256) {
        int r = idx >> 7, c = idx & 127;
        comb[r * SA + c]       = e_t[(row0 + r) * 128 + c];
        comb[r * SA + 192 + c] = prev_h[(row0 + r) * 128 + c];
        xs[r * SX + c]         = x[(row0 + r) * 128 + c];
    }
    for (int idx = tid; idx < 16 * 64; idx += 256) {
        int r = idx >> 6, c = idx & 63;
        comb[r * SA + 128 + c] = r_t[(row0 + r) * 64 + c];
    }
    __syncthreads();

    // ---- GEMM1: two accumulators (gate + candidate) sharing the A fragment ----
    v8f cg = {};
    v8f ch = {};
    for (int k0 = 0; k0 < 320; k0 += 4) {
        const int ak = k0 + 2 * half;            // ISA A layout: halves split K
        v2f a, bq, bh;
        a.x = comb[lrow * SA + ak];
        a.y = comb[lrow * SA + ak + 1];
        const float* pq = Wq_w + ak * 128 + colbase + lrow;   // B: lanes stripe N
        bq.x = pq[0];
        bq.y = pq[128];
        const float* ph = Wh_w + ak * 128 + colbase + lrow;
        bh.x = ph[0];
        bh.y = ph[128];
        cg = __builtin_amdgcn_wmma_f32_16x16x4_f32(false, a, false, bq, (short)0, cg, false, false);
        ch = __builtin_amdgcn_wmma_f32_16x16x4_f32(false, a, false, bh, (short)0, ch, false, false);
    }
    {
        const float bq = Wq_b[colbase + lrow];
        const float bh = Wh_b[colbase + lrow];
#pragma unroll
        for (int r = 0; r < 8; ++r) {
            const int m = r + 8 * half;          // C layout: VGPR r, half selects M+8
            float g  = 1.0f / (1.0f + __expf(-(cg[r] + bq)));
            float cd = tanhf(ch[r] + bh);
            float pv = comb[m * SA + 192 + colbase + lrow];   // prev_h
            ht[m * SH + colbase + lrow] = (1.0f - g) * pv + g * cd;
        }
    }
    __syncthreads();

    // ---- GEMM2: y = sigmoid([h_t | e_t | x] @ Wp + bp) ----
    v8f cy = {};
    for (int k0 = 0; k0 < 384; k0 += 4) {
        const int ak = k0 + 2 * half;
        v2f a, b;
        if (k0 < 128) {
            a.x = ht[lrow * SH + ak];
            a.y = ht[lrow * SH + ak + 1];
        } else if (k0 < 256) {
            a.x = comb[lrow * SA + (ak - 128)];   // e_t block
            a.y = comb[lrow * SA + (ak - 127)];
        } else {
            a.x = xs[lrow * SX + (ak - 256)];     // x block
            a.y = xs[lrow * SX + (ak - 255)];
        }
        const float* pp = Wp_w + ak * 128 + colbase + lrow;
        b.x = pp[0];
        b.y = pp[128];
        cy = __builtin_amdgcn_wmma_f32_16x16x4_f32(false, a, false, b, (short)0, cy, false, false);
    }
    {
        const float bp = Wp_b[colbase + lrow];
#pragma unroll
        for (int r = 0; r < 8; ++r) {
            const int m = r + 8 * half;
            out[(row0 + m) * 128 + colbase + lrow] =
                1.0f / (1.0f + __expf(-(cy[r] + bp)));
        }
    }
}

extern "C" void kernel_launch(void* const* d_in, const int* in_sizes, int n_in,
                              void* d_out, int out_size, void* d_ws, size_t ws_size,
                              hipStream_t stream) {
    (void)in_sizes; (void)n_in; (void)out_size; (void)d_ws; (void)ws_size;
    // setup_inputs() order:
    // 0:x 1:adj 2:e_t 3:r_t 4:prev_h 5:W_heads 6:a_heads 7:W_out 8:a_out
    // 9:Wq_w 10:Wq_b 11:Wh_w 12:Wh_b 13:Wp_w 14:Wp_b
    const float* x      = (const float*)d_in[0];
    const float* e_t    = (const float*)d_in[2];
    const float* r_t    = (const float*)d_in[3];
    const float* prev_h = (const float*)d_in[4];
    const float* Wq_w   = (const float*)d_in[9];
    const float* Wq_b   = (const float*)d_in[10];
    const float* Wh_w   = (const float*)d_in[11];
    const float* Wh_b   = (const float*)d_in[12];
    const float* Wp_w   = (const float*)d_in[13];
    const float* Wp_b   = (const float*)d_in[14];
    float* out = (float*)d_out;

    dim3 grid(N_ROWS / 16);
    dim3 block(256);
    cekt_fused_kernel<<<grid, block, 0, stream>>>(
        x, e_t, r_t, prev_h, Wq_w, Wq_b, Wh_w, Wh_b, Wp_w, Wp_b, out);
}